// YOLOv2Loss_66443144069215
// MI455X (gfx1250) — compile-verified
//
#include <hip/hip_runtime.h>
#include <hip/hip_bf16.h>

// ---------------------------------------------------------------------------
// YOLOv2 loss for MI455X (gfx1250, wave32).
// Memory-bound (43MB outputs read once); f32 VALU main path with coalesced
// channel loads, LDS-staged targets, and a wave-level WMMA f32 16x16x4
// reduction (all-ones B matrix => row sums) for the final scalar.
// ---------------------------------------------------------------------------

#define NB   32      // batch
#define NA   5       // anchors
#define NC   20      // classes
#define NS   52      // grid size
#define NSS  (NS*NS) // 2704
#define NM   50      // max targets
#define NCH  (NA*(5+NC)) // 125

#define IGNORE_THRESH 0.75f
#define OBJ_SCALE     5.0f

#define NLOC_BLOCKS   ((NA*NSS + 255) / 256)        // 53
#define NPARTS        (NB * NLOC_BLOCKS)            // 1696
#define NPARTS_PAD    (((NPARTS + 63) / 64) * 64)   // 1728 (27 WMMA steps)

typedef __attribute__((ext_vector_type(2))) float v2f;
typedef __attribute__((ext_vector_type(8))) float v8f;

__device__ __forceinline__ float sigmoidf(float v) {
  return 1.0f / (1.0f + __expf(-v));
}

// ---------------------------------------------------------------------------
// Kernel A: per-target preprocessing (B*M = 1600 threads).
// Produces: gtbox[b][m][4] (scaled by S), delta[b][m][4],
//           meta[b][m][4] = {cell, best_anchor, class, valid}
// Also zeroes the WMMA pad tail of the partials array so the final
// reduction loop is branch-free (EXEC all-ones, as WMMA requires).
// ---------------------------------------------------------------------------
__global__ __launch_bounds__(128)
void yolo_prep_targets(const float* __restrict__ targets,
                       const float* __restrict__ anchors,
                       float* __restrict__ gtbox,
                       float* __restrict__ tdelta,
                       int*   __restrict__ meta,
                       float* __restrict__ partials) {
  int i = blockIdx.x * blockDim.x + threadIdx.x;

  // zero the padding region [NPARTS, NPARTS_PAD)
  if (i < NPARTS_PAD - NPARTS) partials[NPARTS + i] = 0.0f;

  if (i >= NB * NM) return;

  const float* t = targets + (size_t)i * 6;
  float t0 = t[0], t1 = t[1], t2 = t[2], t3 = t[3], t4 = t[4], t5 = t[5];
  int valid = (t0 + t1 + t2 + t3 + t4 + t5) > 0.0f ? 1 : 0;

  float gx = t0 * (float)NS, gy = t1 * (float)NS;
  float gw = t2 * (float)NS, gh = t3 * (float)NS;
  int gcls = (int)t5;

  float fx = floorf(gx - gw * 0.5f);
  float fy = floorf(gy - gh * 0.5f);
  int cx = (int)fminf(fmaxf(fx, 0.0f), (float)(NS - 1));
  int cy = (int)fminf(fmaxf(fy, 0.0f), (float)(NS - 1));
  int cell = cy * NS + cx;
  float acx = (float)cx + 0.5f;
  float acy = (float)cy + 0.5f;

  // best anchor by IOU at the anchor cell center (first max wins, as argmax)
  int best = 0;
  float bestv = -1.0f;
  float gl = gx - gw * 0.5f, gr = gx + gw * 0.5f;
  float gt_ = gy - gh * 0.5f, gb = gy + gh * 0.5f;
  #pragma unroll
  for (int a = 0; a < NA; ++a) {
    float aw = anchors[a * 2 + 0];
    float ah = anchors[a * 2 + 1];
    float tlx = fmaxf(acx - aw * 0.5f, gl);
    float brx = fminf(acx + aw * 0.5f, gr);
    float tly = fmaxf(acy - ah * 0.5f, gt_);
    float bry = fminf(acy + ah * 0.5f, gb);
    float inter = 0.0f;
    if (tlx < brx && tly < bry) inter = (brx - tlx) * (bry - tly);
    float aiou = inter / (aw * ah + gw * gh - inter);
    if (aiou > bestv) { bestv = aiou; best = a; }
  }
  float aw = anchors[best * 2 + 0];
  float ah = anchors[best * 2 + 1];

  gtbox[i * 4 + 0] = gx;
  gtbox[i * 4 + 1] = gy;
  gtbox[i * 4 + 2] = gw;
  gtbox[i * 4 + 3] = gh;
  tdelta[i * 4 + 0] = gx - acx;
  tdelta[i * 4 + 1] = gy - acy;
  tdelta[i * 4 + 2] = gw / aw;
  tdelta[i * 4 + 3] = gh / ah;
  meta[i * 4 + 0] = cell;
  meta[i * 4 + 1] = best;
  meta[i * 4 + 2] = gcls;
  meta[i * 4 + 3] = valid;
}

// ---------------------------------------------------------------------------
// Kernel B: main loss. grid = (53, B), block = 256.
// Lane mapping: lid = a*S*S + cell -> lanes in a wave share anchor `a` and
// have consecutive cells, so each channel load is a contiguous 128B burst.
// ---------------------------------------------------------------------------
__global__ __launch_bounds__(256)
void yolo_main_loss(const float* __restrict__ outputs,
                    const float* __restrict__ anchors,
                    const float* __restrict__ gtbox,
                    const float* __restrict__ tdelta,
                    const int*   __restrict__ meta,
                    float* __restrict__ partials) {
  __shared__ float s_gt[NM][4];
  __shared__ float s_dl[NM][4];
  __shared__ int   s_cell[NM];
  __shared__ int   s_best[NM];
  __shared__ int   s_cls[NM];
  __shared__ int   s_valid[NM];
  __shared__ float s_anch[NA * 2];
  __shared__ float s_red[256];

  const int b   = blockIdx.y;
  const int tid = threadIdx.x;
  const int lid = blockIdx.x * 256 + tid;   // a*NSS + cell, < NA*NSS = 13520

  if (tid < NM) {
    int gi = b * NM + tid;
    s_gt[tid][0] = gtbox[gi * 4 + 0];
    s_gt[tid][1] = gtbox[gi * 4 + 1];
    s_gt[tid][2] = gtbox[gi * 4 + 2];
    s_gt[tid][3] = gtbox[gi * 4 + 3];
    s_dl[tid][0] = tdelta[gi * 4 + 0];
    s_dl[tid][1] = tdelta[gi * 4 + 1];
    s_dl[tid][2] = tdelta[gi * 4 + 2];
    s_dl[tid][3] = tdelta[gi * 4 + 3];
    s_cell[tid]  = meta[gi * 4 + 0];
    s_best[tid]  = meta[gi * 4 + 1];
    s_cls[tid]   = meta[gi * 4 + 2];
    s_valid[tid] = meta[gi * 4 + 3];
  }
  if (tid < NA * 2) s_anch[tid] = anchors[tid];
  __syncthreads();

  float lsum = 0.0f;
  if (lid < NA * NSS) {
    const int a    = lid / NSS;
    const int cell = lid - a * NSS;
    const int y    = cell / NS;
    const int x    = cell - y * NS;

    // 25 channel values for this location: outputs[b][a*25+c][y][x]
    const float* op = outputs + ((size_t)b * NCH + (size_t)a * 25) * NSS + cell;
    float t[25];
    #pragma unroll
    for (int c = 0; c < 25; ++c) t[c] = op[(size_t)c * NSS];

    const float aw = s_anch[a * 2 + 0];
    const float ah = s_anch[a * 2 + 1];
    const float sig0 = sigmoidf(t[0]);
    const float sig1 = sigmoidf(t[1]);
    const float e2   = __expf(t[2]);
    const float e3   = __expf(t[3]);
    const float pconf = sigmoidf(t[4]);

    const float px = sig0 + (float)x;
    const float py = sig1 + (float)y;
    const float pw = e2 * aw;
    const float ph = e3 * ah;
    const float pl = px - pw * 0.5f, pr = px + pw * 0.5f;
    const float pt = py - ph * 0.5f, pb = py + ph * 0.5f;
    const float area_p = pw * ph;

    // max IOU against all GT boxes (invalid ones contribute 0)
    float maxi = 0.0f;
    #pragma unroll 5
    for (int m = 0; m < NM; ++m) {
      float gx = s_gt[m][0], gy = s_gt[m][1];
      float gw = s_gt[m][2], gh = s_gt[m][3];
      float tlx = fmaxf(pl, gx - gw * 0.5f);
      float brx = fminf(pr, gx + gw * 0.5f);
      float tly = fmaxf(pt, gy - gh * 0.5f);
      float bry = fminf(pb, gy + gh * 0.5f);
      float inter = 0.0f;
      if (tlx < brx && tly < bry) inter = (brx - tlx) * (bry - tly);
      float iou = inter / (area_p + gw * gh - inter);
      if (!s_valid[m]) iou = 0.0f;
      maxi = fmaxf(maxi, iou);
    }

    // find the (last) target assigned to this (cell, anchor) location
    int mm = -1;
    #pragma unroll 5
    for (int m = 0; m < NM; ++m) {
      if (s_valid[m] && s_cell[m] == cell && s_best[m] == a) mm = m;
    }

    // log-softmax normalizer over 20 class logits
    float mv = t[5];
    #pragma unroll
    for (int c = 1; c < NC; ++c) mv = fmaxf(mv, t[5 + c]);
    float se = 0.0f;
    #pragma unroll
    for (int c = 0; c < NC; ++c) se += __expf(t[5 + c] - mv);
    const float logZ = mv + __logf(se);

    const float iou_mask = (maxi >= IGNORE_THRESH) ? 0.0f : 1.0f;
    const float inv2B = 1.0f / (2.0f * (float)NB);
    const float invB  = 1.0f / (float)NB;

    if (mm >= 0) {
      float d0 = sig0 - s_dl[mm][0];
      float d1 = sig1 - s_dl[mm][1];
      float d2 = e2   - s_dl[mm][2];
      float d3 = e3   - s_dl[mm][3];
      float box = d0 * d0 + d1 * d1 + d2 * d2 + d3 * d3;
      float dv  = OBJ_SCALE * (pconf - maxi);   // iou_target == own max_iou
      float nll = -(t[5 + s_cls[mm]] - logZ);
      lsum = box * inv2B + dv * dv * inv2B + nll * invB;
    } else {
      float dv = iou_mask * pconf;
      lsum = dv * dv * inv2B;
    }
  }

  // block tree reduction
  s_red[tid] = lsum;
  __syncthreads();
  #pragma unroll
  for (int off = 128; off > 0; off >>= 1) {
    if (tid < off) s_red[tid] += s_red[tid + off];
    __syncthreads();
  }
  if (tid == 0) partials[blockIdx.y * gridDim.x + blockIdx.x] = s_red[0];
}

// ---------------------------------------------------------------------------
// Kernel C: single-wave final reduction using V_WMMA_F32_16X16X4_F32.
// Partials are padded with zeros to a multiple of 64, so the loop is
// branch-free (EXEC stays all-ones, as WMMA requires). Each lane pulls one
// float2 (global_load_b64) per step; B = all-ones => D[m][n] = sum_k A[m][k],
// so total = column-0 sums held by lanes 0 and 16.
// ---------------------------------------------------------------------------
__global__ __launch_bounds__(32)
void yolo_wmma_reduce(const float* __restrict__ partials, int n_pad,
                      float* __restrict__ out) {
  const int lane = threadIdx.x;
  const v2f* __restrict__ p2 = (const v2f*)partials;   // 8-byte aligned in ws
  v8f acc = {0.0f, 0.0f, 0.0f, 0.0f, 0.0f, 0.0f, 0.0f, 0.0f};
  v2f ones = {1.0f, 1.0f};

  const int steps = n_pad >> 6;   // 64 floats (32 float2) per WMMA
  for (int s = 0; s < steps; ++s) {
    v2f a = p2[s * 32 + lane];
    // D = A(16x4) * ones(4x16) + C   -> row sums replicated across columns
    acc = __builtin_amdgcn_wmma_f32_16x16x4_f32(
        /*neg_a=*/false, a, /*neg_b=*/false, ones,
        /*c_mod=*/(short)0, acc, /*reuse_a=*/false, /*reuse_b=*/false);
  }

  // each lane sums its 8 C registers (rows 0-7 for lanes 0-15, rows 8-15 for
  // lanes 16-31, column lane%16); all columns identical => total = l0 + l16.
  float s = acc[0] + acc[1] + acc[2] + acc[3] +
            acc[4] + acc[5] + acc[6] + acc[7];
  float s16 = __shfl(s, 16, 32);
  if (lane == 0) out[0] = s + s16;
}

// ---------------------------------------------------------------------------
// Host launcher
// ---------------------------------------------------------------------------
extern "C" void kernel_launch(void* const* d_in, const int* in_sizes, int n_in,
                              void* d_out, int out_size, void* d_ws, size_t ws_size,
                              hipStream_t stream) {
  const float* outputs = (const float*)d_in[0];  // (B, 125, S, S) f32
  const float* targets = (const float*)d_in[1];  // (B, M, 6) f32
  const float* anchors = (const float*)d_in[2];  // (5, 2) f32
  float* out = (float*)d_out;

  // workspace partition (all segments 16B-aligned)
  float* gtbox   = (float*)d_ws;                 // B*M*4 floats
  float* tdelta  = gtbox + NB * NM * 4;          // B*M*4 floats
  int*   meta    = (int*)(tdelta + NB * NM * 4); // B*M*4 ints
  float* parts   = (float*)(meta + NB * NM * 4); // NPARTS_PAD floats

  yolo_prep_targets<<<(NB * NM + 127) / 128, 128, 0, stream>>>(
      targets, anchors, gtbox, tdelta, meta, parts);

  dim3 grid(NLOC_BLOCKS, NB);
  yolo_main_loss<<<grid, 256, 0, stream>>>(
      outputs, anchors, gtbox, tdelta, meta, parts);

  yolo_wmma_reduce<<<1, 32, 0, stream>>>(parts, NPARTS_PAD, out);
}